// GDCModel_9414568312944
// MI455X (gfx1250) — compile-verified
//
#include <hip/hip_runtime.h>

#define CIN   128
#define HID   256
#define COUT  40
#define ALPHA 0.05f
#define KSTEPS 10
#define MLP_NODES 32

typedef float v2f __attribute__((ext_vector_type(2)));
typedef float v8f __attribute__((ext_vector_type(8)));

// ---------------- setup kernels ----------------

__global__ __launch_bounds__(256) void k_zero_u32(unsigned* __restrict__ p, int n) {
  int i = blockIdx.x * blockDim.x + threadIdx.x;
  if (i < n) p[i] = 0u;
}

__global__ __launch_bounds__(256) void k_deg(const int* __restrict__ src,
                                             unsigned* __restrict__ deg, int E) {
  int e = blockIdx.x * blockDim.x + threadIdx.x;
  if (e < E) atomicAdd(&deg[src[e]], 1u);
}

// Single-workgroup exclusive scan of deg -> rowstart; also deg_inv = 1/(deg+1).
__global__ __launch_bounds__(1024) void k_scan(const unsigned* __restrict__ deg,
                                               unsigned* __restrict__ rowstart,
                                               unsigned* __restrict__ cursor,
                                               float* __restrict__ dinv, int n) {
  __shared__ unsigned s[1024];
  __shared__ unsigned carry;
  const int tid = threadIdx.x;
  if (tid == 0) carry = 0u;
  __syncthreads();
  for (int base = 0; base < n; base += 1024) {
    const int i = base + tid;
    const unsigned v = (i < n) ? deg[i] : 0u;
    s[tid] = v;
    __syncthreads();
    for (int offs = 1; offs < 1024; offs <<= 1) {
      unsigned t = (tid >= offs) ? s[tid - offs] : 0u;
      __syncthreads();
      s[tid] += t;
      __syncthreads();
    }
    const unsigned excl = carry + s[tid] - v;
    if (i < n) {
      rowstart[i] = excl;
      cursor[i]   = excl;
      dinv[i]     = 1.0f / (float)(v + 1u);   // +1 = self loop; always > 0
    }
    __syncthreads();
    if (tid == 1023) carry += s[1023];
    __syncthreads();
  }
  if (tid == 0) rowstart[n] = carry;          // == E
}

// Bucket-fill CSR adjacency (order within a row is irrelevant for a sum).
__global__ __launch_bounds__(256) void k_fill_csr(const int* __restrict__ src,
                                                  const int* __restrict__ dst,
                                                  unsigned* __restrict__ cursor,
                                                  int* __restrict__ csr_dst, int E) {
  int e = blockIdx.x * blockDim.x + threadIdx.x;
  if (e < E) {
    unsigned pos = atomicAdd(&cursor[src[e]], 1u);
    csr_dst[pos] = dst[e];
  }
}

__global__ __launch_bounds__(256) void k_init(const float* __restrict__ x,
                                              float* __restrict__ z,
                                              float* __restrict__ acc, long n4) {
  long i = (long)blockIdx.x * blockDim.x + threadIdx.x;
  if (i >= n4) return;
  float4 v = ((const float4*)x)[i];
  ((float4*)z)[i] = v;
  ((float4*)acc)[i] = make_float4(ALPHA * v.x, ALPHA * v.y, ALPHA * v.z, ALPHA * v.w);
}

// ---------------- fused gather propagation (no atomics) ----------------
// One wave per node; each edge is one coalesced 512B row read of z_old
// (L2-resident). Inner loop is 4-edge batched so 4 independent b128 gathers
// are in flight per wave (clause + single wait) instead of a serial
// shfl->load->wait chain. Self loop, deg^-1 scale and acc += theta*z fused.
__global__ __launch_bounds__(256) void k_prop(const float* __restrict__ zold,
                                              float* __restrict__ znew,
                                              const unsigned* __restrict__ rowstart,
                                              const float* __restrict__ dinv,
                                              const int* __restrict__ csr_dst,
                                              float* __restrict__ acc,
                                              float theta, int n) {
  const int node = (int)(((long)blockIdx.x * blockDim.x + threadIdx.x) >> 5);
  const int lane = threadIdx.x & 31;
  if (node >= n) return;

  const unsigned beg = rowstart[node];
  const unsigned end = rowstart[node + 1];
  const float* zl = zold + lane * 4;       // lane's 16B slice within any row

  // self-loop contribution
  float4 sum = *(const float4*)(zl + (size_t)node * CIN);

  for (unsigned j = beg; j < end; j += 32) {
    const int mydst = (j + (unsigned)lane < end) ? csr_dst[j + lane] : 0;
    const int cnt = (int)min(32u, end - j);
    int jj = 0;
    for (; jj + 4 <= cnt; jj += 4) {
      const int d0 = __shfl(mydst, jj + 0, 32);
      const int d1 = __shfl(mydst, jj + 1, 32);
      const int d2 = __shfl(mydst, jj + 2, 32);
      const int d3 = __shfl(mydst, jj + 3, 32);
      const float4 v0 = *(const float4*)(zl + (size_t)d0 * CIN);
      const float4 v1 = *(const float4*)(zl + (size_t)d1 * CIN);
      const float4 v2 = *(const float4*)(zl + (size_t)d2 * CIN);
      const float4 v3 = *(const float4*)(zl + (size_t)d3 * CIN);
      sum.x += v0.x; sum.y += v0.y; sum.z += v0.z; sum.w += v0.w;
      sum.x += v1.x; sum.y += v1.y; sum.z += v1.z; sum.w += v1.w;
      sum.x += v2.x; sum.y += v2.y; sum.z += v2.z; sum.w += v2.w;
      sum.x += v3.x; sum.y += v3.y; sum.z += v3.z; sum.w += v3.w;
    }
    for (; jj < cnt; ++jj) {
      const int d = __shfl(mydst, jj, 32);
      const float4 v = *(const float4*)(zl + (size_t)d * CIN);
      sum.x += v.x; sum.y += v.y; sum.z += v.z; sum.w += v.w;
    }
  }

  const float di = dinv[node];
  sum.x *= di; sum.y *= di; sum.z *= di; sum.w *= di;

  *(float4*)(znew + (size_t)node * CIN + lane * 4) = sum;
  float4* ap = (float4*)(acc + (size_t)node * CIN + lane * 4);
  float4 a = *ap;
  a.x += theta * sum.x; a.y += theta * sum.y;
  a.z += theta * sum.z; a.w += theta * sum.w;
  *ap = a;
}

// ---------------- fused MLP + log_softmax using V_WMMA_F32_16X16X4_F32 ----------------

__global__ __launch_bounds__(256) void k_mlp(const float* __restrict__ acc,
                                             const float* __restrict__ W1,
                                             const float* __restrict__ b1,
                                             const float* __restrict__ W2,
                                             const float* __restrict__ b2,
                                             float* __restrict__ out, int nNodes) {
  __shared__ float sA[MLP_NODES][CIN];   // 16 KB
  __shared__ float sH[MLP_NODES][HID];   // 32 KB
  __shared__ float sO[MLP_NODES][48];    //  6 KB

  const int tid = threadIdx.x;
  const int wave = tid >> 5;
  const int lane = tid & 31;
  const int row = lane & 15;      // M (A/D) or N (B) index
  const int khalf = lane >> 4;    // which K-pair this lane holds
  const int node0 = blockIdx.x * MLP_NODES;

  for (int i = tid; i < MLP_NODES * (CIN / 4); i += 256) {
    const int n = i >> 5;
    const int c4 = i & 31;
    const int gn = min(node0 + n, nNodes - 1);
    ((float4*)&sA[n][0])[c4] = ((const float4*)(acc + (size_t)gn * CIN))[c4];
  }
  __syncthreads();

  // ---- GEMM1: h = relu(acc @ W1^T + b1); 32 tiles, 4 per wave ----
  for (int t = wave; t < 32; t += 8) {
    const int mi = t >> 4;
    const int jt = t & 15;
    v8f c = {0.f, 0.f, 0.f, 0.f, 0.f, 0.f, 0.f, 0.f};
    const float* Ar = &sA[mi * 16 + row][0];
    const float* Br = W1 + (size_t)(jt * 16 + row) * CIN;   // B[k][col]=W1[col][k]
#pragma unroll 8
    for (int k = 0; k < CIN / 4; ++k) {
      v2f a, b;
      a.x = Ar[k * 4 + khalf * 2 + 0];
      a.y = Ar[k * 4 + khalf * 2 + 1];
      b.x = Br[k * 4 + khalf * 2 + 0];
      b.y = Br[k * 4 + khalf * 2 + 1];
      c = __builtin_amdgcn_wmma_f32_16x16x4_f32(false, a, false, b,
                                                (short)0, c, false, false);
    }
    const int j = jt * 16 + row;
    const float bj = b1[j];
#pragma unroll
    for (int r = 0; r < 8; ++r) {
      const float v = c[r] + bj;
      sH[mi * 16 + r + 8 * khalf][j] = v > 0.f ? v : 0.f;
    }
  }
  __syncthreads();

  // ---- GEMM2: out = h @ W2^T + b2; 6 tiles on waves 0..5 ----
  if (wave < 6) {
    const int mi = (wave >= 3) ? 1 : 0;
    const int ot = (wave >= 3) ? wave - 3 : wave;
    v8f c = {0.f, 0.f, 0.f, 0.f, 0.f, 0.f, 0.f, 0.f};
    const int oc = min(ot * 16 + row, COUT - 1);   // clamp padded cols
    const float* Ar = &sH[mi * 16 + row][0];
    const float* Br = W2 + (size_t)oc * HID;
#pragma unroll 8
    for (int k = 0; k < HID / 4; ++k) {
      v2f a, b;
      a.x = Ar[k * 4 + khalf * 2 + 0];
      a.y = Ar[k * 4 + khalf * 2 + 1];
      b.x = Br[k * 4 + khalf * 2 + 0];
      b.y = Br[k * 4 + khalf * 2 + 1];
      c = __builtin_amdgcn_wmma_f32_16x16x4_f32(false, a, false, b,
                                                (short)0, c, false, false);
    }
    const int o = ot * 16 + row;
    const float bo = b2[min(o, COUT - 1)];
#pragma unroll
    for (int r = 0; r < 8; ++r)
      sO[mi * 16 + r + 8 * khalf][o] = c[r] + bo;
  }
  __syncthreads();

  // ---- log_softmax over 40 valid columns, one thread per node ----
  if (tid < MLP_NODES) {
    const int gn = node0 + tid;
    if (gn < nNodes) {
      float mx = sO[tid][0];
      for (int o = 1; o < COUT; ++o) mx = fmaxf(mx, sO[tid][o]);
      float s = 0.f;
      for (int o = 0; o < COUT; ++o) s += __expf(sO[tid][o] - mx);
      const float lse = __logf(s) + mx;
      float* op = out + (size_t)gn * COUT;
      for (int o = 0; o < COUT; ++o) op[o] = sO[tid][o] - lse;
    }
  }
}

// ---------------- launch ----------------

extern "C" void kernel_launch(void* const* d_in, const int* in_sizes, int n_in,
                              void* d_out, int out_size, void* d_ws, size_t ws_size,
                              hipStream_t stream) {
  const float* x  = (const float*)d_in[0];
  const int*   ei = (const int*)d_in[1];
  const float* W1 = (const float*)d_in[3];
  const float* b1 = (const float*)d_in[4];
  const float* W2 = (const float*)d_in[5];
  const float* b2 = (const float*)d_in[6];
  float* out = (float*)d_out;

  const int N = in_sizes[0] / CIN;
  const int E = in_sizes[1] / 2;
  const int* src = ei;        // edge_index[0]
  const int* dst = ei + E;    // edge_index[1]

  size_t off = 0;
  auto carve = [&](size_t bytes) -> void* {
    void* p = (char*)d_ws + off;
    off += (bytes + 255) & ~(size_t)255;
    return p;
  };
  unsigned* deg      = (unsigned*)carve((size_t)N * sizeof(unsigned));
  unsigned* rowstart = (unsigned*)carve((size_t)(N + 1) * sizeof(unsigned));
  unsigned* cursor   = (unsigned*)carve((size_t)N * sizeof(unsigned));
  float*    dinv     = (float*)carve((size_t)N * sizeof(float));
  int*      csr_dst  = (int*)carve((size_t)E * sizeof(int));
  float*    za       = (float*)carve((size_t)N * CIN * sizeof(float));
  float*    zb       = (float*)carve((size_t)N * CIN * sizeof(float));
  float*    accp     = (float*)carve((size_t)N * CIN * sizeof(float));

  const int  B  = 256;
  const long n4 = (long)N * CIN / 4;

  // one-time CSR build
  k_zero_u32<<<(N + B - 1) / B, B, 0, stream>>>(deg, N);
  k_deg<<<(E + B - 1) / B, B, 0, stream>>>(src, deg, E);
  k_scan<<<1, 1024, 0, stream>>>(deg, rowstart, cursor, dinv, N);
  k_fill_csr<<<(E + B - 1) / B, B, 0, stream>>>(src, dst, cursor, csr_dst, E);

  k_init<<<(int)((n4 + B - 1) / B), B, 0, stream>>>(x, za, accp, n4);

  float theta = ALPHA;
  for (int k = 1; k <= KSTEPS; ++k) {
    theta *= (1.0f - ALPHA);
    const long waves = (long)N * 32;
    k_prop<<<(int)((waves + B - 1) / B), B, 0, stream>>>(za, zb, rowstart, dinv,
                                                         csr_dst, accp, theta, N);
    float* tmp = za; za = zb; zb = tmp;
  }

  k_mlp<<<(N + MLP_NODES - 1) / MLP_NODES, B, 0, stream>>>(accp, W1, b1, W2, b2, out, N);
}